// GuidedCrossAttention_10316511445681
// MI455X (gfx1250) — compile-verified
//
#include <hip/hip_runtime.h>
#include <hip/hip_bf16.h>

#define DCH 1024            // model dim D
#define NROWS 65536         // B*G*N = 16 * 4096 rows

typedef __attribute__((ext_vector_type(2))) float v2f;
typedef __attribute__((ext_vector_type(4))) float v4f;
typedef __attribute__((ext_vector_type(8))) float v8f;

// ---------------------------------------------------------------------------
// OUT[16, 1024] = IN[16, 1024] @ W^T + bias   (W is [1024,1024], row-major)
// One wave per 16-column tile of OUT; 64 tiles total -> 8 blocks x 8 waves.
// Uses V_WMMA_F32_16X16X4_F32 (full f32 precision, K-loop of 256 steps).
// ---------------------------------------------------------------------------
__global__ __launch_bounds__(256) void gca_gemm16_wmma(
    const float* __restrict__ IN, const float* __restrict__ W,
    const float* __restrict__ bias, float* __restrict__ OUT) {
  const int lane = threadIdx.x & 31;
  const int wave = threadIdx.x >> 5;
  const int tile = blockIdx.x * 8 + wave;      // 0..63
  const int c0   = tile * 16;                  // output column base

  const int mn   = lane & 15;                  // row m for A frag, col n for B frag
  const int kbo  = (lane >> 4) << 1;           // +0 for lanes 0-15, +2 for lanes 16-31

  const float* arow = IN + (size_t)mn * DCH;            // A: row m of IN
  const float* brow = W  + (size_t)(c0 + mn) * DCH;     // B: row (c0+n) of W (== col of W^T)

  v8f acc = {};                                // f32 16x16 accumulator (8 VGPRs)
  #pragma unroll 4
  for (int k0 = 0; k0 < DCH; k0 += 4) {
    const v2f av = *(const v2f*)(arow + k0 + kbo);
    const v2f bv = *(const v2f*)(brow + k0 + kbo);
    acc = __builtin_amdgcn_wmma_f32_16x16x4_f32(
        /*neg_a=*/false, av, /*neg_b=*/false, bv,
        /*c_mod=*/(short)0, acc, /*reuse_a=*/false, /*reuse_b=*/false);
  }

  // C/D layout: VGPR r -> M = r + 8*(lane>=16); N = lane&15
  const int mbase = (lane >> 4) * 8;
  const float bval = bias[c0 + mn];
  #pragma unroll
  for (int r = 0; r < 8; ++r) {
    OUT[(size_t)(mbase + r) * DCH + c0 + mn] = acc[r] + bval;
  }
}

// ---------------------------------------------------------------------------
// y[row,:] = rmsnorm(x[row,:] + o[row>>12,:]) * w     (row length 1024)
// One wave per row; x read once (nontemporal), y written once (nontemporal).
// Pure HBM-bandwidth pass: 512 MB total -> ~22 us at 23.3 TB/s.
// ---------------------------------------------------------------------------
__global__ __launch_bounds__(256) void gca_addrms(
    const float* __restrict__ x, const float* __restrict__ o,
    const float* __restrict__ w, float* __restrict__ y) {
  const int lane = threadIdx.x & 31;
  const int wave = threadIdx.x >> 5;
  const int row  = blockIdx.x * 8 + wave;      // 0..65535
  const int bg   = row >> 12;                  // row / 4096

  const float* xr = x + (size_t)row * DCH;
  const float* orow = o + (size_t)bg * DCH;
  float* yr = y + (size_t)row * DCH;

  v4f t[8];
  float ss = 0.0f;
  #pragma unroll
  for (int c = 0; c < 8; ++c) {
    const int idx = c * 128 + lane * 4;
    const v4f xv = __builtin_nontemporal_load((const v4f*)(xr + idx));
    const v4f ov = *(const v4f*)(orow + idx);   // cached (reused 4096x per bg)
    const v4f tv = xv + ov;
    t[c] = tv;
    ss = fmaf(tv.x, tv.x, ss);
    ss = fmaf(tv.y, tv.y, ss);
    ss = fmaf(tv.z, tv.z, ss);
    ss = fmaf(tv.w, tv.w, ss);
  }

  // wave32 allreduce of sum of squares
  #pragma unroll
  for (int m = 16; m >= 1; m >>= 1) ss += __shfl_xor(ss, m, 32);

  const float inv = rsqrtf(ss * (1.0f / (float)DCH) + 1e-6f);

  #pragma unroll
  for (int c = 0; c < 8; ++c) {
    const int idx = c * 128 + lane * 4;
    const v4f wv = *(const v4f*)(w + idx);      // cached
    const v4f ov = t[c] * inv * wv;
    __builtin_nontemporal_store(ov, (v4f*)(yr + idx));
  }
}

extern "C" void kernel_launch(void* const* d_in, const int* in_sizes, int n_in,
                              void* d_out, int out_size, void* d_ws, size_t ws_size,
                              hipStream_t stream) {
  (void)in_sizes; (void)n_in; (void)out_size; (void)ws_size;
  const float* x        = (const float*)d_in[0];   // [2,8,4096,1024]
  const float* guidance = (const float*)d_in[1];   // [2,8,1024] -> [16,1024]
  const float* in_w     = (const float*)d_in[2];   // [3072,1024]
  const float* in_b     = (const float*)d_in[3];   // [3072]
  const float* out_w    = (const float*)d_in[4];   // [1024,1024]
  const float* out_b    = (const float*)d_in[5];   // [1024]
  const float* norm_w   = (const float*)d_in[6];   // [1024]
  float* y = (float*)d_out;

  // Workspace: V[16,1024] then O[16,1024] (64 KB each)
  float* V = (float*)d_ws;
  float* O = V + 16 * DCH;

  const float* Wv = in_w + (size_t)2 * DCH * DCH;  // rows [2D, 3D) of in_proj_w
  const float* bv = in_b + 2 * DCH;

  // kv_len == 1 => softmax == 1 => attention output == V broadcast over N.
  gca_gemm16_wmma<<<8, 256, 0, stream>>>(guidance, Wv, bv, V);   // V = g @ Wv^T + bv
  gca_gemm16_wmma<<<8, 256, 0, stream>>>(V, out_w, out_b, O);    // O = V @ Wo^T + bo
  gca_addrms<<<NROWS / 8, 256, 0, stream>>>(x, O, norm_w, y);    // y = rmsnorm(x+O)*w
}